// LinearConv2D_31112743092858
// MI455X (gfx1250) — compile-verified
//
#include <hip/hip_runtime.h>

// CDNA5 / gfx1250, wave32. Fused grouped linear-conv + box-sum + leaky-relu
// using V_WMMA_F32_16X16X4_F32 (exact fp32 matrix path).

typedef __attribute__((ext_vector_type(2))) float v2f;
typedef __attribute__((ext_vector_type(8))) float v8f;

#define B_     8
#define CH_    128
#define F_IN   32
#define T_IN   1024
#define G_     8
#define D_     16
#define N_     16
#define W_     16
#define KS_    4
#define T_OUT  257     // (1024+16-16)/4 + 1
#define F_OUT  17      // (32+2-2)/2 + 1

#define XS_STRIDE   80            // 76 used, padded to 80 floats (even, bank-friendly)
#define XS_PER_WAVE (D_ * XS_STRIDE)

__global__ __launch_bounds__(256)
void linconv2d_wmma_f32(const float* __restrict__ x,
                        const float* __restrict__ weight,
                        float* __restrict__ out) {
  // [fs][n][d][w] weight pair slab, zero-filled outside f-range: 32 KB
  __shared__ __align__(16) float wlds[2 * N_ * D_ * W_];
  // per-wave x window slab [d][lt]: 8 waves * 5 KB = 40 KB
  __shared__ __align__(16) float xs[8 * XS_PER_WAVE];

  const int tid  = threadIdx.x;
  const int wv   = tid >> 5;      // wave id 0..7
  const int lane = tid & 31;
  const int hi   = lane >> 4;     // 0 for lanes 0-15, 1 for 16-31
  const int lo   = lane & 15;

  const int fo = blockIdx.y;          // 0..16
  const int z  = blockIdx.z;          // b*8 + g
  const int bi = z >> 3;
  const int g  = z & 7;

  // ---- stage weight pair into LDS (block-cooperative, zero pad f=-1 / f=32) ----
  for (int idx = tid; idx < 2 * N_ * D_ * W_; idx += 256) {
    const int fs  = idx >> 12;            // /4096
    const int rem = idx & 4095;
    const int n   = rem >> 8;             // /256
    const int d   = (rem >> 4) & 15;
    const int w   = rem & 15;
    const int f   = 2 * fo - 1 + fs;
    float v = 0.0f;
    if (f >= 0 && f < F_IN)
      v = weight[(size_t)(g * N_ + n) * (D_ * F_IN * W_) + d * (F_IN * W_) + f * W_ + w];
    wlds[idx] = v;
  }
  __syncthreads();

  const int tile = blockIdx.x * 8 + wv;   // 16-row t tile
  const int t0   = tile * 16;
  if (t0 >= T_OUT) return;                // ragged tail waves idle (after the only barrier)

  float* myxs = &xs[wv * XS_PER_WAVE];
  v8f acc = {};

  // Accumulate the GEMM over both f values feeding this f_out window.
  for (int fs = 0; fs < 2; ++fs) {
    const int f  = 2 * fo - 1 + fs;
    const int fc = f < 0 ? 0 : (f > F_IN - 1 ? F_IN - 1 : f); // clamp: weights are 0 there anyway

    // stage x slab: d rows, local padded time [0,76)  ->  time = t0*4 + lt - 8
    for (int idx = lane; idx < D_ * 76; idx += 32) {
      const int d  = idx / 76;
      const int lt = idx - d * 76;
      const int ot = t0 * KS_ + lt - 8;
      float v = 0.0f;
      if (ot >= 0 && ot < T_IN)
        v = x[(((size_t)bi * CH_ + g * D_ + d) * F_IN + fc) * T_IN + ot];
      myxs[d * XS_STRIDE + lt] = v;
    }
    // same-wave DS ops are in order; compiler inserts s_wait_dscnt as needed

    // K = (d, w) = 256, consumed 4-at-a-time by V_WMMA_F32_16X16X4_F32.
    // A frag (16x4 f32): lane lo = row m; v0/v1 hold K = 2*hi, 2*hi+1.
    // B frag (4x16 f32): lane lo = col n; same K split (mirror of A layout).
    #pragma unroll
    for (int d = 0; d < D_; ++d) {
      #pragma unroll
      for (int wq = 0; wq < 4; ++wq) {
        const int kk = wq * 4 + 2 * hi;
        v2f a = *(const v2f*)&myxs[d * XS_STRIDE + lo * KS_ + kk];
        v2f b = *(const v2f*)&wlds[fs * 4096 + lo * 256 + d * 16 + kk];
        acc = __builtin_amdgcn_wmma_f32_16x16x4_f32(
            /*neg_a=*/false, a, /*neg_b=*/false, b,
            /*c_mod=*/(short)0, acc, /*reuse_a=*/false, /*reuse_b=*/false);
      }
    }
  }

  // ---- epilogue: leaky relu + LDS transpose for t-contiguous stores ----
  // C layout: VGPR j, lane: M = j + 8*hi (t row), N = lo (n).  tr[n][m], stride 17.
  #pragma unroll
  for (int j = 0; j < 8; ++j) {
    float vv = acc[j];
    vv = vv > 0.0f ? vv : 0.01f * vv;
    myxs[lo * 17 + j + 8 * hi] = vv;
  }
  // each lane writes 8 consecutive t for one n -> coalesced 32B runs
  const int nn = lane >> 1;
  const int tb = (lane & 1) * 8;
  const size_t ob = (((size_t)bi * CH_ + g * N_ + nn) * F_OUT + fo) * (size_t)T_OUT + t0 + tb;
  #pragma unroll
  for (int e = 0; e < 8; ++e) {
    const int tg = t0 + tb + e;
    if (tg < T_OUT) out[ob + e] = myxs[nn * 17 + tb + e];
  }
}

extern "C" void kernel_launch(void* const* d_in, const int* in_sizes, int n_in,
                              void* d_out, int out_size, void* d_ws, size_t ws_size,
                              hipStream_t stream) {
  (void)in_sizes; (void)n_in; (void)out_size; (void)d_ws; (void)ws_size;
  const float* x = (const float*)d_in[0];
  const float* w = (const float*)d_in[1];
  float* out     = (float*)d_out;

  // grid: x = t-tile blocks (ceil(17 tiles / 8 waves) = 3), y = f_out (17), z = b*g (64)
  dim3 grid(3, F_OUT, B_ * G_);
  linconv2d_wmma_f32<<<grid, 256, 0, stream>>>(x, w, out);
}